// Node2Vec_60653528154240
// MI455X (gfx1250) — compile-verified
//
#include <hip/hip_runtime.h>

typedef __attribute__((ext_vector_type(2))) float v2f;
typedef __attribute__((ext_vector_type(8))) float v8f;

#define DIM 128
#define ROW_LEN 128
#define BATCH 8192
#define WPB 4   // waves per block

// One wave32 computes one batch row:
//   scores[l] = dot(X[rt[b,l]], X[rt[b,0]])  for l = 0..127
// via V_WMMA_F32_16X16X4_F32 with B = x0-chunk broadcast across N,
// then loss[b] = m*logsumexp(scores) - sum(scores[1..m]).
//
// Roofline: 8192*128 gathered rows x 512B = 512MB logical traffic, but X
// (51.2MB) stays resident in the 192MB L2, so this is an L2-gather-bound
// kernel (~tens of us floor at 23.3TB/s-class bandwidth). f32 WMMA keeps
// numerics exact while using the CDNA5 matrix pipe.
__global__ __launch_bounds__(32 * WPB) void node2vec_loss_kernel(
    const int* __restrict__ rt, const float* __restrict__ X,
    const int* __restrict__ mptr, float* __restrict__ loss_out)
{
    __shared__ float lds_x0[WPB][DIM];
    __shared__ float lds_sc[WPB][ROW_LEN];

    const int lane = threadIdx.x & 31;
    const int wave = threadIdx.x >> 5;
    const int b    = blockIdx.x * WPB + wave;

    const int*   idx  = rt + (size_t)b * ROW_LEN;
    const int    row0 = idx[0];
    const float* x0p  = X + (size_t)row0 * DIM;

    // Preload x0 (128 floats) into this wave's LDS slice: 32 lanes x float4.
    {
        float4 v = *(const float4*)(x0p + lane * 4);
        *(float4*)(&lds_x0[wave][lane * 4]) = v;
    }
    __syncthreads();

    // WMMA A-layout (16x4 f32): lane<16 holds K={0,1}, lane>=16 holds K={2,3}.
    const int mrow = lane & 15;
    const int koff = (lane >> 4) << 1;   // 0 or 2

    for (int tile = 0; tile < 8; ++tile) {
        const int    l    = tile * 16 + mrow;
        const int    rowl = idx[l];
        const float* ap   = X + (size_t)rowl * DIM + koff;   // gathered A row slice
        const float* bp   = &lds_x0[wave][koff];             // broadcast B source

        v8f acc = {0.f, 0.f, 0.f, 0.f, 0.f, 0.f, 0.f, 0.f};
        #pragma unroll
        for (int k0 = 0; k0 < DIM; k0 += 4) {
            v2f a, bb;
            a[0]  = ap[k0];
            a[1]  = ap[k0 + 1];
            bb[0] = bp[k0];
            bb[1] = bp[k0 + 1];
            // D = A(16x4) * B(4x16) + C ; all N columns of B identical, so
            // column 0 of D accumulates the 16 dot products.
            acc = __builtin_amdgcn_wmma_f32_16x16x4_f32(
                false, a, false, bb, (short)0, acc, false, false);
        }
        // D layout: lane 0 vgpr j -> (M=j, N=0); lane 16 vgpr j -> (M=8+j, N=0).
        if (lane == 0) {
            #pragma unroll
            for (int j = 0; j < 8; ++j) lds_sc[wave][tile * 16 + j] = acc[j];
        } else if (lane == 16) {
            #pragma unroll
            for (int j = 0; j < 8; ++j) lds_sc[wave][tile * 16 + 8 + j] = acc[j];
        }
    }
    __syncthreads();

    // In-wave reduction: each lane owns scores[lane*4 .. lane*4+3].
    const int mval = mptr[0];
    float s[4];
    #pragma unroll
    for (int j = 0; j < 4; ++j) s[j] = lds_sc[wave][lane * 4 + j];

    float mx = fmaxf(fmaxf(s[0], s[1]), fmaxf(s[2], s[3]));
    #pragma unroll
    for (int off = 16; off >= 1; off >>= 1)
        mx = fmaxf(mx, __shfl_xor(mx, off, 32));

    float se = 0.f;
    #pragma unroll
    for (int j = 0; j < 4; ++j) se += expf(s[j] - mx);
    #pragma unroll
    for (int off = 16; off >= 1; off >>= 1)
        se += __shfl_xor(se, off, 32);

    float pos = 0.f;
    #pragma unroll
    for (int j = 0; j < 4; ++j) {
        const int i = lane * 4 + j;
        if (i >= 1 && i <= mval) pos += s[j];
    }
    #pragma unroll
    for (int off = 16; off >= 1; off >>= 1)
        pos += __shfl_xor(pos, off, 32);

    if (lane == 0) {
        const float lse = mx + logf(se);
        loss_out[b] = (float)mval * lse - pos;
    }
}

// Deterministic single-block mean over the 8192 per-row losses.
__global__ __launch_bounds__(256) void reduce_mean_kernel(
    const float* __restrict__ part, float* __restrict__ out, int n)
{
    __shared__ float sm[256];
    float s = 0.f;
    for (int i = threadIdx.x; i < n; i += 256) s += part[i];
    sm[threadIdx.x] = s;
    __syncthreads();
    #pragma unroll
    for (int off = 128; off > 0; off >>= 1) {
        if ((int)threadIdx.x < off) sm[threadIdx.x] += sm[threadIdx.x + off];
        __syncthreads();
    }
    if (threadIdx.x == 0) out[0] = sm[0] / (float)n;
}

extern "C" void kernel_launch(void* const* d_in, const int* in_sizes, int n_in,
                              void* d_out, int out_size, void* d_ws, size_t ws_size,
                              hipStream_t stream) {
    const int*   rt = (const int*)d_in[0];     // (BATCH, ROW_LEN) int32
    const float* X  = (const float*)d_in[1];   // (N_NODES, DIM) f32
    const int*   m  = (const int*)d_in[2];     // scalar
    float* part = (float*)d_ws;                // BATCH floats of scratch
    float* out  = (float*)d_out;

    node2vec_loss_kernel<<<BATCH / WPB, 32 * WPB, 0, stream>>>(rt, X, m, part);
    reduce_mean_kernel<<<1, 256, 0, stream>>>(part, out, BATCH);
}